// MGNN_76201309765701
// MI455X (gfx1250) — compile-verified
//
#include <hip/hip_runtime.h>
#include <hip/hip_bf16.h>
#include <math.h>

// ---------------------------------------------------------------------------
// MGNN forward for MI455X (gfx1250). fp32 WMMA (V_WMMA_F32_16X16X4_F32) for all
// large GEMMs with the A-tile staged into LDS via CDNA5 async global->LDS
// copies (GLOBAL_LOAD_ASYNC_TO_LDS_B128 + s_wait_asynccnt); f32 global atomics
// for graph aggregation; tiny kernels for the 10/5-cluster hierarchical
// pooling (gumbel-hard == argmax in the forward pass).
// ---------------------------------------------------------------------------

typedef __attribute__((ext_vector_type(2))) float v2f;
typedef __attribute__((ext_vector_type(8))) float v8f;

#define NFEAT 128
#define NHID  128
#define MAX_K 512
#define LDS_PAD 4   // +4 floats/row: 16B-aligned rows, bank step 4 (16 rows -> 16 banks)

// ---------------------------------------------------------------------------
// Generic fp32 WMMA GEMM: C[n x M] = A[n x K] @ W[K x M] (+bias) (optional relu)
// Block = 4 waves = 16 rows x 64 cols of C. The 16 x K A tile is staged once
// per block into LDS with async copies and shared by all 4 waves; each wave
// computes one 16x16 tile, iterating K in steps of 4 with f32 WMMA.
// Requirements: n % 16 == 0, K % 32 == 0, M % 64 == 0, 16B-aligned A.
// No divergence anywhere -> EXEC all ones (WMMA + async-load requirement).
// ---------------------------------------------------------------------------
__global__ __launch_bounds__(128) void wmma_gemm_f32(
    const float* __restrict__ A, int lda,
    const float* __restrict__ W, int ldw,
    const float* __restrict__ bias,
    float* __restrict__ C, int ldc,
    int K, int do_relu)
{
    __shared__ float As[16 * (MAX_K + LDS_PAD)];
    const int kpad = K + LDS_PAD;
    const int tid  = threadIdx.x;
    const int lane = tid & 31;
    const int wave = tid >> 5;
    const int row0 = blockIdx.x * 16;
    const int col0 = (blockIdx.y * 4 + wave) * 16;

    // --- stage A tile (16 x K) into LDS with CDNA5 async global->LDS b128 ---
    const int nchunk = (16 * K) / 4;               // float4 chunks, % 128 == 0
    for (int ch = tid; ch < nchunk; ch += 128) {
        int r = (ch * 4) / K;                      // K % 4 == 0: no row straddle
        int c = (ch * 4) % K;
        const float* gsrc = A + (size_t)(row0 + r) * lda + c;
        unsigned ldst = (unsigned)(unsigned long long)(uintptr_t)&As[r * kpad + c];
        asm volatile("global_load_async_to_lds_b128 %0, %1, off"
                     :: "v"(ldst), "v"(gsrc) : "memory");
    }
    asm volatile("s_wait_asynccnt 0x0" ::: "memory");
    __syncthreads();

    const int half = lane >> 4;   // 0: lanes 0-15 carry K {0,1}; 1: lanes 16-31 carry K {2,3}
    const int l    = lane & 15;   // A: row within tile; B/C: col within tile

    v8f acc = {};
    const float* arow = As + l * kpad + half * 2;                  // LDS (ds_load_b64)
    const float* bcol = W + (size_t)(half * 2) * ldw + col0 + l;   // global, coalesced

    for (int k = 0; k < K; k += 4) {
        v2f a, b;
        a.x = arow[k];                     // A[row][k + 2*half]
        a.y = arow[k + 1];                 // A[row][k + 2*half + 1]
        b.x = bcol[(size_t)k * ldw];       // W[k + 2*half][col]
        b.y = bcol[(size_t)(k + 1) * ldw]; // W[k + 2*half + 1][col]
        acc = __builtin_amdgcn_wmma_f32_16x16x4_f32(
            false, a, false, b, (short)0, acc, false, false);
    }

    const float bb = bias ? bias[col0 + l] : 0.0f;
    #pragma unroll
    for (int v = 0; v < 8; ++v) {
        float x = acc[v] + bb;
        if (do_relu) x = fmaxf(x, 0.0f);
        C[(size_t)(row0 + v + half * 8) * ldc + (col0 + l)] = x;
    }
}

// ------------------------- graph aggregation -------------------------------
__global__ void deg_accum(const int* __restrict__ col, const float* __restrict__ ew,
                          float* __restrict__ deg, int E)
{
    int e = blockIdx.x * blockDim.x + threadIdx.x;
    if (e < E) atomicAdd(&deg[col[e]], ew[e]);
}

__global__ void deg_finish(const float* __restrict__ deg, float* __restrict__ dinv, int n)
{
    int i = blockIdx.x * blockDim.x + threadIdx.x;
    if (i < n) dinv[i] = rsqrtf(deg[i] + 1.0f);
}

// agg[col] += (dinv[row]*ew*dinv[col]) * h[row][:]
__global__ void gcn_scatter(const int* __restrict__ row, const int* __restrict__ col,
                            const float* __restrict__ ew, const float* __restrict__ dinv,
                            const float* __restrict__ h, float* __restrict__ agg,
                            long total)
{
    long idx = (long)blockIdx.x * blockDim.x + threadIdx.x;
    if (idx >= total) return;
    int e = (int)(idx >> 7);
    int c = (int)(idx & 127);
    int r = row[e], cl = col[e];
    float nrm = dinv[r] * ew[e] * dinv[cl];
    atomicAdd(&agg[(size_t)cl * NHID + c], nrm * h[(size_t)r * NHID + c]);
}

// h = relu(agg + h*dinv^2 + bias)   (in place on h)
__global__ void gcn_combine(float* __restrict__ h, const float* __restrict__ agg,
                            const float* __restrict__ dinv, const float* __restrict__ bias,
                            long total)
{
    long idx = (long)blockIdx.x * blockDim.x + threadIdx.x;
    if (idx >= total) return;
    int i = (int)(idx >> 7);
    int c = (int)(idx & 127);
    float d = dinv[i];
    float v = agg[idx] + h[idx] * d * d + bias[c];
    h[idx] = fmaxf(v, 0.0f);
}

// ------------------------------- batchnorm ---------------------------------
__global__ __launch_bounds__(256) void bn_stats(const float* __restrict__ x,
                                                float* __restrict__ mean,
                                                float* __restrict__ rstd, int n)
{
    int c = blockIdx.x;     // one block per feature column (128 blocks)
    __shared__ float s1[256], s2[256];
    float a = 0.f, b = 0.f;
    for (int i = threadIdx.x; i < n; i += blockDim.x) {
        float v = x[(size_t)i * NHID + c];
        a += v; b += v * v;
    }
    s1[threadIdx.x] = a; s2[threadIdx.x] = b;
    __syncthreads();
    for (int s = 128; s > 0; s >>= 1) {
        if (threadIdx.x < s) {
            s1[threadIdx.x] += s1[threadIdx.x + s];
            s2[threadIdx.x] += s2[threadIdx.x + s];
        }
        __syncthreads();
    }
    if (threadIdx.x == 0) {
        float m = s1[0] / n;
        float v = s2[0] / n - m * m;
        mean[c] = m;
        rstd[c] = rsqrtf(v + 1e-5f);
    }
}

__global__ void bn_apply(const float* __restrict__ x, const float* __restrict__ mean,
                         const float* __restrict__ rstd, const float* __restrict__ g,
                         const float* __restrict__ b, float* __restrict__ out, long total)
{
    long idx = (long)blockIdx.x * blockDim.x + threadIdx.x;
    if (idx >= total) return;
    int c = (int)(idx & 127);
    out[idx] = (x[idx] - mean[c]) * rstd[c] * g[c] + b[c];
}

// ------------------------------ concat helpers -----------------------------
__global__ void build_cat(const float* __restrict__ x, const float* __restrict__ h1,
                          const float* __restrict__ h2, float* __restrict__ cat, long total)
{
    long idx = (long)blockIdx.x * blockDim.x + threadIdx.x;
    if (idx >= total) return;
    int i = (int)(idx >> 7);
    int c = (int)(idx & 127);
    float* r = cat + (size_t)i * 384;
    r[c]       = x[idx];
    r[128 + c] = h1[idx];
    r[256 + c] = h2[idx];
}

__global__ void build_rep(const float* __restrict__ bottom, const float* __restrict__ lat0,
                          const float* __restrict__ lat1, const int* __restrict__ c0,
                          const int* __restrict__ c1, float* __restrict__ rep, long total)
{
    long idx = (long)blockIdx.x * blockDim.x + threadIdx.x;
    if (idx >= total) return;
    int i = (int)(idx >> 7);
    int c = (int)(idx & 127);
    int a = c0[i];
    int b = c1[a];
    float* r = rep + (size_t)i * 384;
    r[c]       = bottom[idx];
    r[128 + c] = lat0[a * NHID + c];
    r[256 + c] = lat1[b * NHID + c];
}

// ------------------------------ pooling level 0 ----------------------------
// cluster0[i] = argmax_c ( bottom[i]@ml0_w + ml0_b + gumbel0 )[c]   (C0=10)
__global__ void assign_level0(const float* __restrict__ bottom, const float* __restrict__ mlw,
                              const float* __restrict__ mlb, const float* __restrict__ gum,
                              int* __restrict__ c0, int n)
{
    int i = blockIdx.x * blockDim.x + threadIdx.x;
    if (i >= n) return;
    float lg[10];
    for (int c = 0; c < 10; ++c) lg[c] = mlb[c] + gum[(size_t)i * 10 + c];
    const float* xr = bottom + (size_t)i * NHID;
    for (int k = 0; k < NHID; ++k) {
        float xv = xr[k];
        #pragma unroll
        for (int c = 0; c < 10; ++c) lg[c] += xv * mlw[k * 10 + c];
    }
    float best = lg[0]; int bi = 0;
    for (int c = 1; c < 10; ++c) if (lg[c] > best) { best = lg[c]; bi = c; }
    c0[i] = bi;
}

// xp[c][col] = sum over nodes in cluster c of latent[i][col]. One block per column.
__global__ __launch_bounds__(256) void xp_accum(const float* __restrict__ latent,
                                                const int* __restrict__ cidx,
                                                float* __restrict__ xp, int n, int C)
{
    int col = blockIdx.x;
    float acc[10];
    for (int j = 0; j < 10; ++j) acc[j] = 0.f;
    for (int i = threadIdx.x; i < n; i += blockDim.x)
        acc[cidx[i]] += latent[(size_t)i * NHID + col];
    __shared__ float s[10 * 256];
    for (int j = 0; j < C; ++j) s[j * 256 + threadIdx.x] = acc[j];
    __syncthreads();
    for (int st = 128; st > 0; st >>= 1) {
        if (threadIdx.x < st)
            for (int j = 0; j < C; ++j)
                s[j * 256 + threadIdx.x] += s[j * 256 + threadIdx.x + st];
        __syncthreads();
    }
    if (threadIdx.x < C) xp[threadIdx.x * NHID + col] = s[threadIdx.x * 256];
}

// adjp[a][b] += sum_e ew[e] [c0[row]=a][c0[col]=b]  (10x10 histogram, LDS-staged)
__global__ __launch_bounds__(256) void adj_pool_edges(const int* __restrict__ row,
                                                      const int* __restrict__ col,
                                                      const float* __restrict__ ew,
                                                      const int* __restrict__ c0,
                                                      float* __restrict__ adjp, int E)
{
    __shared__ float s[100];
    if (threadIdx.x < 100) s[threadIdx.x] = 0.f;
    __syncthreads();
    int e = blockIdx.x * blockDim.x + threadIdx.x;
    if (e < E) atomicAdd(&s[c0[row[e]] * 10 + c0[col[e]]], ew[e]);
    __syncthreads();
    if (threadIdx.x < 100 && s[threadIdx.x] != 0.f)
        atomicAdd(&adjp[threadIdx.x], s[threadIdx.x]);
}

// Single block: adjn = adj/sum(adj); xpn = l2rows(xp); t = adjn@xpn;
// latent = tanh(t @ me_w + me_b). C <= 10.
__global__ __launch_bounds__(256) void pool_level(const float* __restrict__ xp_raw,
                                                  const float* __restrict__ adj_raw,
                                                  const float* __restrict__ mew,
                                                  const float* __restrict__ meb,
                                                  float* __restrict__ latent,
                                                  float* __restrict__ adjn_out, int C)
{
    __shared__ float xpn[10 * 128];
    __shared__ float t[10 * 128];
    __shared__ float adjn[100];
    __shared__ float rn[10];
    __shared__ float sinv;
    int tid = threadIdx.x;
    if (tid == 0) {
        float s = 0.f;
        for (int i = 0; i < C * C; ++i) s += adj_raw[i];
        sinv = (s != 0.f) ? 1.f / s : 0.f;
    }
    if (tid < C) {
        float s = 0.f;
        for (int k = 0; k < NHID; ++k) { float v = xp_raw[tid * NHID + k]; s += v * v; }
        rn[tid] = 1.f / fmaxf(sqrtf(s), 1e-12f);
    }
    __syncthreads();
    for (int i = tid; i < C * C; i += blockDim.x) {
        adjn[i] = adj_raw[i] * sinv;
        adjn_out[i] = adjn[i];
    }
    for (int i = tid; i < C * NHID; i += blockDim.x)
        xpn[i] = xp_raw[i] * rn[i >> 7];
    __syncthreads();
    for (int i = tid; i < C * NHID; i += blockDim.x) {
        int r = i >> 7, c = i & 127;
        float s = 0.f;
        for (int j = 0; j < C; ++j) s += adjn[r * C + j] * xpn[j * NHID + c];
        t[i] = s;
    }
    __syncthreads();
    for (int i = tid; i < C * NHID; i += blockDim.x) {
        int r = i >> 7, c = i & 127;
        float s = meb[c];
        for (int k = 0; k < NHID; ++k) s += t[r * NHID + k] * mew[k * NHID + c];
        latent[i] = tanhf(s);
    }
}

// ------------------------------ pooling level 1 ----------------------------
// Single block: cluster1[r]=argmax(latent0@ml1_w+b+g1); xp1 = seg-sum(latent0);
// adjp1 = pooled adjn0.  (C0=10 -> C1=5)
__global__ __launch_bounds__(128) void level1_prep(const float* __restrict__ lat0,
                                                   const float* __restrict__ ml1w,
                                                   const float* __restrict__ ml1b,
                                                   const float* __restrict__ gum1,
                                                   const float* __restrict__ adjn0,
                                                   int* __restrict__ c1,
                                                   float* __restrict__ xp1,
                                                   float* __restrict__ adjp1)
{
    __shared__ int c1s[10];
    int tid = threadIdx.x;
    if (tid < 10) {
        float lg[5];
        for (int c = 0; c < 5; ++c) lg[c] = ml1b[c] + gum1[tid * 5 + c];
        for (int k = 0; k < NHID; ++k) {
            float v = lat0[tid * NHID + k];
            #pragma unroll
            for (int c = 0; c < 5; ++c) lg[c] += v * ml1w[k * 5 + c];
        }
        float best = lg[0]; int bi = 0;
        for (int c = 1; c < 5; ++c) if (lg[c] > best) { best = lg[c]; bi = c; }
        c1s[tid] = bi;
        c1[tid] = bi;
    }
    __syncthreads();
    if (tid < 128) {
        float acc[5] = {0.f, 0.f, 0.f, 0.f, 0.f};
        for (int r = 0; r < 10; ++r) acc[c1s[r]] += lat0[r * NHID + tid];
        for (int j = 0; j < 5; ++j) xp1[j * NHID + tid] = acc[j];
    }
    if (tid < 25) {
        int a = tid / 5, b = tid % 5;
        float s = 0.f;
        for (int i = 0; i < 10; ++i)
            if (c1s[i] == a)
                for (int j = 0; j < 10; ++j)
                    if (c1s[j] == b) s += adjn0[i * 10 + j];
        adjp1[tid] = s;
    }
}

// ------------------------------- output head -------------------------------
__global__ void fc2_head(const float* __restrict__ a, const float* __restrict__ w,
                         const float* __restrict__ b, float* __restrict__ out, int n)
{
    int i = blockIdx.x * blockDim.x + threadIdx.x;
    if (i >= n) return;
    float s = b[0];
    const float* ar = a + (size_t)i * 512;
    for (int k = 0; k < 512; ++k) s += ar[k] * w[k];
    out[i] = fmaxf(s, 0.0f);
}

// ---------------------------------------------------------------------------
extern "C" void kernel_launch(void* const* d_in, const int* in_sizes, int n_in,
                              void* d_out, int out_size, void* d_ws, size_t ws_size,
                              hipStream_t stream)
{
    const float* x       = (const float*)d_in[0];
    const float* ew      = (const float*)d_in[1];
    const float* gum0    = (const float*)d_in[2];
    const float* gum1    = (const float*)d_in[3];
    const float* conv1_w = (const float*)d_in[4];
    const float* conv1_b = (const float*)d_in[5];
    const float* conv2_w = (const float*)d_in[6];
    const float* conv2_b = (const float*)d_in[7];
    const float* bn1_g   = (const float*)d_in[8];
    const float* bn1_b   = (const float*)d_in[9];
    const float* bn2_g   = (const float*)d_in[10];
    const float* bn2_b   = (const float*)d_in[11];
    const float* enc1_w  = (const float*)d_in[12];
    const float* enc1_b  = (const float*)d_in[13];
    const float* enc2_w  = (const float*)d_in[14];
    const float* enc2_b  = (const float*)d_in[15];
    const float* ml0_w   = (const float*)d_in[16];
    const float* ml0_b   = (const float*)d_in[17];
    const float* ml1_w   = (const float*)d_in[18];
    const float* ml1_b   = (const float*)d_in[19];
    const float* me0_w   = (const float*)d_in[20];
    const float* me0_b   = (const float*)d_in[21];
    const float* me1_w   = (const float*)d_in[22];
    const float* me1_b   = (const float*)d_in[23];
    const float* fc1_w   = (const float*)d_in[24];
    const float* fc1_b   = (const float*)d_in[25];
    const float* fc2_w   = (const float*)d_in[26];
    const float* fc2_b   = (const float*)d_in[27];
    const int*   eidx    = (const int*)d_in[28];

    const int N = in_sizes[0] / NFEAT;   // 10000
    const int E = in_sizes[1];           // 320000
    const int* erow = eidx;
    const int* ecol = eidx + E;

    // ---- workspace carve-up (floats) ----
    float* ws = (float*)d_ws;
    size_t off = 0;
    float* hbuf   = ws + off; off += (size_t)N * NHID;
    float* agg    = ws + off; off += (size_t)N * NHID;
    float* h1     = ws + off; off += (size_t)N * NHID;
    float* h2     = ws + off; off += (size_t)N * NHID;
    float* catbuf = ws + off; off += (size_t)N * 384;   // also reused as `rep`
    float* enc_t  = ws + off; off += (size_t)N * NHID;
    float* bottom = ws + off; off += (size_t)N * NHID;
    float* buf512 = ws + off; off += (size_t)N * 512;
    float* deg    = ws + off; off += N;
    float* dinv   = ws + off; off += N;
    float* bn_m   = ws + off; off += 128;
    float* bn_r   = ws + off; off += 128;
    float* xp0    = ws + off; off += 10 * NHID;
    float* lat0   = ws + off; off += 10 * NHID;
    float* adjp0  = ws + off; off += 100;
    float* adjn0  = ws + off; off += 100;
    float* xp1    = ws + off; off += 5 * NHID;
    float* lat1   = ws + off; off += 5 * NHID;
    float* adjp1  = ws + off; off += 25;
    float* adjn1  = ws + off; off += 25;
    int*   c0     = (int*)(ws + off); off += N;
    int*   c1     = (int*)(ws + off); off += 16;

    const long totNH = (long)N * NHID;
    const dim3 blk256(256);
    const dim3 grdNH((unsigned)((totNH + 255) / 256));
    const dim3 grdE((unsigned)((E + 255) / 256));
    const dim3 grdEF((unsigned)(((long)E * NHID + 255) / 256));
    const dim3 grdN((unsigned)((N + 255) / 256));

    auto gemm = [&](const float* A, int lda, const float* W, int ldw,
                    const float* bias, float* C, int ldc, int K, int M, int relu) {
        dim3 grid(N / 16, (M / 16) / 4);
        wmma_gemm_f32<<<grid, 128, 0, stream>>>(A, lda, W, ldw, bias, C, ldc, K, relu);
    };

    // ---- degree / symmetric norm (shared by both convs) ----
    hipMemsetAsync(deg, 0, (size_t)N * sizeof(float), stream);
    deg_accum<<<grdE, blk256, 0, stream>>>(ecol, ew, deg, E);
    deg_finish<<<grdN, blk256, 0, stream>>>(deg, dinv, N);

    // ---- GCN conv1 ----
    gemm(x, NFEAT, conv1_w, NHID, nullptr, hbuf, NHID, NFEAT, NHID, 0);
    hipMemsetAsync(agg, 0, totNH * sizeof(float), stream);
    gcn_scatter<<<grdEF, blk256, 0, stream>>>(erow, ecol, ew, dinv, hbuf, agg, (long)E * NHID);
    gcn_combine<<<grdNH, blk256, 0, stream>>>(hbuf, agg, dinv, conv1_b, totNH);
    bn_stats<<<128, blk256, 0, stream>>>(hbuf, bn_m, bn_r, N);
    bn_apply<<<grdNH, blk256, 0, stream>>>(hbuf, bn_m, bn_r, bn1_g, bn1_b, h1, totNH);

    // ---- GCN conv2 ----
    gemm(h1, NHID, conv2_w, NHID, nullptr, hbuf, NHID, NHID, NHID, 0);
    hipMemsetAsync(agg, 0, totNH * sizeof(float), stream);
    gcn_scatter<<<grdEF, blk256, 0, stream>>>(erow, ecol, ew, dinv, hbuf, agg, (long)E * NHID);
    gcn_combine<<<grdNH, blk256, 0, stream>>>(hbuf, agg, dinv, conv2_b, totNH);
    bn_stats<<<128, blk256, 0, stream>>>(hbuf, bn_m, bn_r, N);
    bn_apply<<<grdNH, blk256, 0, stream>>>(hbuf, bn_m, bn_r, bn2_g, bn2_b, h2, totNH);

    // ---- encoder MLP ----
    build_cat<<<grdNH, blk256, 0, stream>>>(x, h1, h2, catbuf, totNH);
    gemm(catbuf, 384, enc1_w, NHID, enc1_b, enc_t, NHID, 384, NHID, 1);
    gemm(enc_t, NHID, enc2_w, NHID, enc2_b, bottom, NHID, NHID, NHID, 1);

    // ---- pooling level 0 (C0 = 10) ----
    assign_level0<<<grdN, blk256, 0, stream>>>(bottom, ml0_w, ml0_b, gum0, c0, N);
    xp_accum<<<128, blk256, 0, stream>>>(bottom, c0, xp0, N, 10);
    hipMemsetAsync(adjp0, 0, 100 * sizeof(float), stream);
    adj_pool_edges<<<grdE, blk256, 0, stream>>>(erow, ecol, ew, c0, adjp0, E);
    pool_level<<<1, blk256, 0, stream>>>(xp0, adjp0, me0_w, me0_b, lat0, adjn0, 10);

    // ---- pooling level 1 (C1 = 5) ----
    level1_prep<<<1, 128, 0, stream>>>(lat0, ml1_w, ml1_b, gum1, adjn0, c1, xp1, adjp1);
    pool_level<<<1, blk256, 0, stream>>>(xp1, adjp1, me1_w, me1_b, lat1, adjn1, 5);

    // ---- readout head ----
    build_rep<<<grdNH, blk256, 0, stream>>>(bottom, lat0, lat1, c0, c1, catbuf, totNH);
    gemm(catbuf, 384, fc1_w, 512, fc1_b, buf512, 512, 384, 512, 1);
    fc2_head<<<grdN, blk256, 0, stream>>>(buf512, fc2_w, fc2_b, (float*)d_out, N);
}